// ThermalFDMSolver_75368086110543
// MI455X (gfx1250) — compile-verified
//
#include <hip/hip_runtime.h>
#include <stdint.h>

// ---------------------------------------------------------------------------
// ThermalFDMSolver.step() for MI455X (gfx1250).
//
// ONE workgroup, 256 threads (8 wave32):
//   Phase 1: async global->LDS copy of Tw (8 KB) via
//            GLOBAL_LOAD_ASYNC_TO_LDS_B128  (ASYNCcnt / s_wait_asynccnt).
//   Phase 2: lane 0 runs the sequential gas march entirely in LDS
//            (2047-step dependent chain; 1 v_exp+v_log per step after exact
//             Re*Pr algebraic simplification).
//   Phase 3: all 256 threads do the vectorized wall Euler update, coalesced
//            stores of (Tg_new | Tw_new) to d_out.
// ---------------------------------------------------------------------------

#define NN 2048
#define NT 256

// ---- exact problem constants, evaluated in double at compile time ----
constexpr double PI_D   = 3.14159265358979323846;
constexpr double L_D    = 0.5;
constexpr double ID_D   = 0.004;
constexpr double OD_D   = 0.006;
constexpr double DZ_D   = L_D / NN;
constexpr double ACS_D  = PI_D * ID_D * ID_D / 4.0;
constexpr double AW_D   = PI_D * (OD_D * OD_D - ID_D * ID_D) / 4.0;
constexpr double VW_D   = AW_D * L_D;
constexpr double FM_D   = 1.5e-6;
constexpr double PW_D   = 150.0;
constexpr double DT_D   = 0.01;
constexpr double HAMB_D = 10.0;
constexpr double MWKG_D = 4.0026e-3;
constexpr double CPM_D  = 20.786 / MWKG_D;                      // cp mass [J/kg/K]
// Re*Pr = FM*ID*cp/(A_CS*k)  (rho, v cancel exactly)
// Nu*k  = max(1.86*FM*ID^2*cp/(A_CS*L), 4.36*k);  h = Nu*k/ID
constexpr double C1_D   = 1.86 * FM_D * ID_D * ID_D * CPM_D / (ACS_D * L_D);
constexpr double GASC_D = PI_D * ID_D * DZ_D / (FM_D * CPM_D);  // dTg = GASC*h*(Tw-Tg)
constexpr double QGEN_D = (PW_D / VW_D) * AW_D * DZ_D;
constexpr double CONV_D = PI_D * ID_D * DZ_D;
constexpr double LOSS_D = HAMB_D * PI_D * OD_D * DZ_D;
constexpr double MCELL_D = 7100.0 * AW_D * DZ_D;
constexpr float  LN2_F  = 0.69314718055994531f;

// ---- hardware transcendentals: v_log_f32 (log2), v_exp_f32 (exp2) ----
__device__ __forceinline__ float hw_log2(float x) {
#if __has_builtin(__builtin_amdgcn_logf)
    return __builtin_amdgcn_logf(x);
#else
    return log2f(x);
#endif
}
__device__ __forceinline__ float hw_exp2(float x) {
#if __has_builtin(__builtin_amdgcn_exp2f)
    return __builtin_amdgcn_exp2f(x);
#else
    return exp2f(x);
#endif
}

__device__ __forceinline__ float clampT(float x) {
    return fminf(fmaxf(x, 1.0f), 1500.0f);
}

// k_He(T_film) = 2.682e-3 * (T_film + 273.15)^0.71
__device__ __forceinline__ float k_he(float Tf) {
    return 2.682e-3f * hw_exp2(0.71f * hw_log2(Tf + 273.15f));
}

// h = max(C1, 4.36*k)/ID  (exact rewrite of the reference HTC)
__device__ __forceinline__ float htc_f(float Tg, float Tw) {
    float k = k_he(0.5f * (Tg + Tw));
    return fmaxf((float)C1_D, 4.36f * k) * (float)(1.0 / ID_D);
}

// ---- async global -> LDS, 16 bytes/lane (ASYNCcnt path) ----
typedef int v4i __attribute__((vector_size(16)));
typedef __attribute__((address_space(1))) v4i GV4;
typedef __attribute__((address_space(3))) v4i LV4;
typedef __attribute__((address_space(3))) float LFloat;

__device__ __forceinline__ void async_load16(const float* g, float* l) {
#if __has_builtin(__builtin_amdgcn_global_load_async_to_lds_b128)
    __builtin_amdgcn_global_load_async_to_lds_b128((GV4*)g, (LV4*)l, 0, 0);
#else
    unsigned lds_off = (unsigned)(uintptr_t)(LFloat*)l;   // 32-bit LDS byte addr
    unsigned long long ga = (unsigned long long)(uintptr_t)g;
    asm volatile("global_load_async_to_lds_b128 %0, %1, off"
                 :: "v"(lds_off), "v"(ga) : "memory");
#endif
}

__device__ __forceinline__ void wait_async0() {
#if __has_builtin(__builtin_amdgcn_s_wait_asynccnt)
    __builtin_amdgcn_s_wait_asynccnt(0);
#else
    asm volatile("s_wait_asynccnt 0x0" ::: "memory");
#endif
}

__global__ __launch_bounds__(NT)
void thermal_fdm_step(const float* __restrict__ Tw_in, float* __restrict__ out)
{
    __shared__ float sTw[NN];
    __shared__ float sTg[NN];
    const int t = threadIdx.x;

    // ---- Phase 1: async stage Tw into LDS (2 x b128 per lane = 8 KB) ----
    async_load16(Tw_in + t * 4,          sTw + t * 4);
    async_load16(Tw_in + t * 4 + NT * 4, sTw + t * 4 + NT * 4);
    wait_async0();
    __syncthreads();

    // ---- Phase 2: sequential gas march (one lane, all in LDS) ----
    if (t == 0) {
        float Tg  = 25.0f;                 // clip(TIN) = 25
        sTg[0] = Tg;
        float Twp = clampT(sTw[0]);
        #pragma unroll 4
        for (int i = 1; i < NN; ++i) {
            float Twn = sTw[i];            // prefetch next node's wall temp
            float h   = htc_f(Tg, Twp);
            Tg = Tg + (float)GASC_D * h * (Twp - Tg);
            Tg = clampT(Tg);
            sTg[i] = Tg;
            Twp = clampT(Twn);
        }
    }
    __syncthreads();

    // ---- Phase 3: vectorized wall Forward-Euler update + stores ----
    #pragma unroll
    for (int j = 0; j < NN / NT; ++j) {
        int   i   = t + NT * j;            // coalesced per wave
        float Tw  = clampT(sTw[i]);
        float Tg  = sTg[i];
        float h   = htc_f(Tg, Tw);
        float cpw = (0.1f * (LN2_F * hw_log2(Tw)) + 0.3f) * 1000.0f; // J/kg/K
        float q   = (float)QGEN_D
                  - h * (float)CONV_D * (Tw - Tg)
                  - (float)LOSS_D * (Tw - 25.0f);
        float dTw = q / ((float)MCELL_D * cpw);
        float Twn = clampT(Tw + dTw * (float)DT_D);
        out[i]      = Tg;                  // Tg_new (already clipped)
        out[NN + i] = Twn;                 // Tw_new
    }
}

extern "C" void kernel_launch(void* const* d_in, const int* in_sizes, int n_in,
                              void* d_out, int out_size, void* d_ws, size_t ws_size,
                              hipStream_t stream) {
    (void)in_sizes; (void)n_in; (void)d_ws; (void)ws_size; (void)out_size;
    const float* Tw = (const float*)d_in[0];
    float* out = (float*)d_out;            // [0..2047]=Tg_new, [2048..4095]=Tw_new
    thermal_fdm_step<<<1, NT, 0, stream>>>(Tw, out);
}